// EVEMixtralSparseBlock_46162308497852
// MI455X (gfx1250) — compile-verified
//
#include <hip/hip_runtime.h>
#include <math.h>

// Problem sizes (fixed by the reference)
#define Tt  4096      // B*S tokens
#define Hh  4096      // hidden
#define FFf 14336     // ffn dim
#define LDSW 20       // padded LDS row stride (floats): 80B rows -> 16B-aligned b128 slots, conflict-free frag reads

typedef __attribute__((ext_vector_type(2))) float v2f;
typedef __attribute__((ext_vector_type(8))) float v8f;

__device__ __forceinline__ v8f wmma4(v2f a, v2f b, v8f c) {
#if __has_builtin(__builtin_amdgcn_wmma_f32_16x16x4_f32)
  // (neg_a, A, neg_b, B, c_mod, C, reuse_a, reuse_b)
  return __builtin_amdgcn_wmma_f32_16x16x4_f32(false, a, false, b, (short)0, c, false, false);
#else
  return c;
#endif
}

// LDS byte offset of a __shared__ object: flat LDS aperture keeps the
// allocation offset in addr[31:0] (ISA 10.2 aperture mapping).
__device__ __forceinline__ uint32_t lds_off(const void* p) {
  return (uint32_t)(uintptr_t)p;
}

// Async global -> LDS copy, 16 bytes per lane. Tracked by ASYNCcnt.
__device__ __forceinline__ void g2l_b128(uint32_t lds_addr, const float* gaddr) {
  asm volatile("global_load_async_to_lds_b128 %0, %1, off"
               :: "v"(lds_addr), "v"(gaddr)
               : "memory");
}

__device__ __forceinline__ void wait_async0() {
  asm volatile("s_wait_asynccnt 0x0" ::: "memory");
}
__device__ __forceinline__ void wait_ds0() {
  asm volatile("s_wait_dscnt 0x0" ::: "memory");
}

// ---------------------------------------------------------------------------
// Router: logits = x @ router_w.T + b ; top-2 expert ids. (Top-2 of logits ==
// top-2 of softmax; normalized top-k weights sum to exactly 1, so the shared
// expert output needs no weighting.) One wave per token.
// ---------------------------------------------------------------------------
__global__ __launch_bounds__(256) void router_kernel(
    const float* __restrict__ x, const float* __restrict__ rw,
    const float* __restrict__ rb, float* __restrict__ logits,
    int* __restrict__ sel) {
  const int wv = threadIdx.x >> 5, lane = threadIdx.x & 31;
  const int t = blockIdx.x * 8 + wv;
  const float* xr = x + (size_t)t * Hh;
  float acc[8];
#pragma unroll
  for (int e = 0; e < 8; ++e) acc[e] = 0.f;
  for (int j = lane; j < Hh; j += 32) {
    float xv = xr[j];
#pragma unroll
    for (int e = 0; e < 8; ++e) acc[e] += xv * rw[e * Hh + j];
  }
#pragma unroll
  for (int e = 0; e < 8; ++e) {
#pragma unroll
    for (int off = 16; off > 0; off >>= 1)
      acc[e] += __shfl_xor(acc[e], off, 32);
  }
  if (lane == 0) {
    float v0 = acc[0] + rb[0];
    logits[(size_t)t * 8] = v0;
    float m0 = v0, m1 = -INFINITY;
    int e0 = 0, e1 = 0;
#pragma unroll
    for (int e = 1; e < 8; ++e) {
      float v = acc[e] + rb[e];
      logits[(size_t)t * 8 + e] = v;
      if (v > m0) { m1 = m0; e1 = e0; m0 = v; e0 = e; }
      else if (v > m1) { m1 = v; e1 = e; }
    }
    sel[t] = e0 | (e1 << 8);
  }
}

// ---------------------------------------------------------------------------
// LoRA stage 1: z[t][0..15] = x[t].lora_A[e0][r], z[t][16..31] = e1 rows.
// One wave per token; lane owns one (expert, r) pair. lora_A (2MB) is L2
// resident.
// ---------------------------------------------------------------------------
__global__ __launch_bounds__(256) void lora_z_kernel(
    const float* __restrict__ x, const float* __restrict__ lA,
    const int* __restrict__ sel, float* __restrict__ z) {
  const int wv = threadIdx.x >> 5, lane = threadIdx.x & 31;
  const int t = blockIdx.x * 8 + wv;
  const int sp = sel[t];
  const int e = (lane < 16) ? (sp & 0xff) : ((sp >> 8) & 0xff);
  const int r = lane & 15;
  const float* ar = lA + ((size_t)e * 16 + r) * Hh;   // lora_A [E][R][H]
  const float* xr = x + (size_t)t * Hh;
  float acc = 0.f;
  for (int j = 0; j < Hh; j += 8) {
#pragma unroll
    for (int q = 0; q < 8; ++q) acc += xr[j + q] * ar[j + q];
  }
  z[(size_t)t * 32 + lane] = acc;
}

// ---------------------------------------------------------------------------
// LoRA stage 2: out[t][h] = SCALING * (z0.B[e0][h][:] + z1.B[e1][h][:]).
// Writes the full [T,H] out region (deterministic under graph replay).
// ---------------------------------------------------------------------------
__global__ __launch_bounds__(256) void lora_out_kernel(
    const float* __restrict__ z, const float* __restrict__ lB,
    const int* __restrict__ sel, float* __restrict__ out) {
  const int bx = blockIdx.x;
  const int t = bx >> 4;                       // 16 blocks of 256 cover H=4096
  const int h = ((bx & 15) << 8) + threadIdx.x;
  const int sp = sel[t];
  const int e0 = sp & 0xff, e1 = (sp >> 8) & 0xff;
  const float* zr = z + (size_t)t * 32;
  const float* b0 = lB + ((size_t)e0 * Hh + h) * 16;   // lora_B [E][H][R]
  const float* b1 = lB + ((size_t)e1 * Hh + h) * 16;
  float acc = 0.f;
#pragma unroll
  for (int r = 0; r < 16; ++r) acc += zr[r] * b0[r] + zr[16 + r] * b1[r];
  out[(size_t)t * Hh + h] = 2.0f * acc;        // SCALING = 32/16
}

// ---------------------------------------------------------------------------
// Fused gate+up GEMM: act = silu(x@w1.T) * (x@w3.T).
// Tile 128x128, K-chunk 16, DOUBLE-BUFFERED LDS filled by async global->LDS
// copies (ASYNCcnt) so chunk k+1 streams in while chunk k runs on WMMA.
// 8 waves: 4 along M x 2 along N; per wave 2x4 16x16 f32 accs per matrix.
// ---------------------------------------------------------------------------
__global__ __launch_bounds__(256) void gate_up_kernel(
    const float* __restrict__ x, const float* __restrict__ w1,
    const float* __restrict__ w3, float* __restrict__ act) {
  __shared__ __align__(16) float As[2][128 * LDSW];
  __shared__ __align__(16) float B1s[2][128 * LDSW];
  __shared__ __align__(16) float B3s[2][128 * LDSW];
  const int t0 = blockIdx.y * 128;
  const int n0 = blockIdx.x * 128;
  const int tid = threadIdx.x;
  const int lane = tid & 31;
  const int wv = tid >> 5;
  const int wm = wv & 3;   // wave row (M)
  const int wn = wv >> 2;  // wave col (N)
  const int m16 = lane & 15;
  const int hi = lane >> 4;

  v8f acc1[2][4], acc3[2][4];
#pragma unroll
  for (int i = 0; i < 2; ++i)
#pragma unroll
    for (int j = 0; j < 4; ++j)
#pragma unroll
      for (int q = 0; q < 8; ++q) { acc1[i][j][q] = 0.f; acc3[i][j][q] = 0.f; }

  // Per-thread fixed slots: 2 x b128 per tile (512 slots / 256 threads).
  int row0 = (tid + 0)   >> 2, c0 = ((tid + 0)   & 3) << 2;
  int row1 = (tid + 256) >> 2, c1 = ((tid + 256) & 3) << 2;

  auto issue = [&](int buf, int kc) {
    g2l_b128(lds_off(&As[buf][row0 * LDSW + c0]),  x  + (size_t)(t0 + row0) * Hh + kc + c0);
    g2l_b128(lds_off(&B1s[buf][row0 * LDSW + c0]), w1 + (size_t)(n0 + row0) * Hh + kc + c0);
    g2l_b128(lds_off(&B3s[buf][row0 * LDSW + c0]), w3 + (size_t)(n0 + row0) * Hh + kc + c0);
    g2l_b128(lds_off(&As[buf][row1 * LDSW + c1]),  x  + (size_t)(t0 + row1) * Hh + kc + c1);
    g2l_b128(lds_off(&B1s[buf][row1 * LDSW + c1]), w1 + (size_t)(n0 + row1) * Hh + kc + c1);
    g2l_b128(lds_off(&B3s[buf][row1 * LDSW + c1]), w3 + (size_t)(n0 + row1) * Hh + kc + c1);
  };

  issue(0, 0);
  for (int kc = 0; kc < Hh; kc += 16) {
    const int cur = (kc >> 4) & 1;
    wait_async0();      // my async fills of buf[cur] have landed in LDS
    __syncthreads();    // everyone's have
    if (kc + 16 < Hh) issue(cur ^ 1, kc + 16);  // stream next chunk under compute
#pragma unroll
    for (int ks = 0; ks < 4; ++ks) {
      const int ko = ks << 2;
      // A 16x4 frag: lanes 0-15 M=lane K={ko,ko+1}; lanes 16-31 K={ko+2,ko+3}
      v2f af[2];
#pragma unroll
      for (int ms = 0; ms < 2; ++ms) {
        const float* p = &As[cur][(wm * 32 + ms * 16 + m16) * LDSW + ko + (hi << 1)];
        af[ms].x = p[0]; af[ms].y = p[1];
      }
      // B 4x16 frag: VGPR0 rows K={ko,ko+1}, VGPR1 rows K={ko+2,ko+3}
      v2f bf1[4], bf3[4];
#pragma unroll
      for (int ns = 0; ns < 4; ++ns) {
        const float* p1 = &B1s[cur][(wn * 64 + ns * 16 + m16) * LDSW + ko + hi];
        bf1[ns].x = p1[0]; bf1[ns].y = p1[2];
        const float* p3 = &B3s[cur][(wn * 64 + ns * 16 + m16) * LDSW + ko + hi];
        bf3[ns].x = p3[0]; bf3[ns].y = p3[2];
      }
#pragma unroll
      for (int ms = 0; ms < 2; ++ms)
#pragma unroll
        for (int ns = 0; ns < 4; ++ns) {
          acc1[ms][ns] = wmma4(af[ms], bf1[ns], acc1[ms][ns]);
          acc3[ms][ns] = wmma4(af[ms], bf3[ns], acc3[ms][ns]);
        }
    }
    wait_ds0();         // my LDS reads of buf[cur] are done
    __syncthreads();    // safe for buf[cur] to be overwritten next iteration
  }
  // epilogue: act = silu(h1) * h3   (C/D VGPR i: lanes0-15 M=i, lanes16-31 M=i+8)
#pragma unroll
  for (int ms = 0; ms < 2; ++ms)
#pragma unroll
    for (int ns = 0; ns < 4; ++ns)
#pragma unroll
      for (int i = 0; i < 8; ++i) {
        int row = t0 + wm * 32 + ms * 16 + i + (hi << 3);
        int col = n0 + wn * 64 + ns * 16 + m16;
        float h1 = acc1[ms][ns][i];
        float h3 = acc3[ms][ns][i];
        float s = h1 / (1.0f + __expf(-h1));
        act[(size_t)row * FFf + col] = s * h3;
      }
}

// ---------------------------------------------------------------------------
// Down GEMM: out += act @ w2.T (out already holds the LoRA contribution).
// Same double-buffered async pipeline; K = FF.
// ---------------------------------------------------------------------------
__global__ __launch_bounds__(256) void down_kernel(
    const float* __restrict__ act, const float* __restrict__ w2,
    float* __restrict__ out) {
  __shared__ __align__(16) float As[2][128 * LDSW];
  __shared__ __align__(16) float Bs[2][128 * LDSW];
  const int t0 = blockIdx.y * 128;
  const int n0 = blockIdx.x * 128;
  const int tid = threadIdx.x;
  const int lane = tid & 31;
  const int wv = tid >> 5;
  const int wm = wv & 3;
  const int wn = wv >> 2;
  const int m16 = lane & 15;
  const int hi = lane >> 4;

  v8f acc[2][4];
#pragma unroll
  for (int i = 0; i < 2; ++i)
#pragma unroll
    for (int j = 0; j < 4; ++j)
#pragma unroll
      for (int q = 0; q < 8; ++q) acc[i][j][q] = 0.f;

  int row0 = (tid + 0)   >> 2, c0 = ((tid + 0)   & 3) << 2;
  int row1 = (tid + 256) >> 2, c1 = ((tid + 256) & 3) << 2;

  auto issue = [&](int buf, int kc) {
    g2l_b128(lds_off(&As[buf][row0 * LDSW + c0]), act + (size_t)(t0 + row0) * FFf + kc + c0);
    g2l_b128(lds_off(&Bs[buf][row0 * LDSW + c0]), w2  + (size_t)(n0 + row0) * FFf + kc + c0);
    g2l_b128(lds_off(&As[buf][row1 * LDSW + c1]), act + (size_t)(t0 + row1) * FFf + kc + c1);
    g2l_b128(lds_off(&Bs[buf][row1 * LDSW + c1]), w2  + (size_t)(n0 + row1) * FFf + kc + c1);
  };

  issue(0, 0);
  for (int kc = 0; kc < FFf; kc += 16) {
    const int cur = (kc >> 4) & 1;
    wait_async0();
    __syncthreads();
    if (kc + 16 < FFf) issue(cur ^ 1, kc + 16);
#pragma unroll
    for (int ks = 0; ks < 4; ++ks) {
      const int ko = ks << 2;
      v2f af[2];
#pragma unroll
      for (int ms = 0; ms < 2; ++ms) {
        const float* p = &As[cur][(wm * 32 + ms * 16 + m16) * LDSW + ko + (hi << 1)];
        af[ms].x = p[0]; af[ms].y = p[1];
      }
      v2f bf[4];
#pragma unroll
      for (int ns = 0; ns < 4; ++ns) {
        const float* p = &Bs[cur][(wn * 64 + ns * 16 + m16) * LDSW + ko + hi];
        bf[ns].x = p[0]; bf[ns].y = p[2];
      }
#pragma unroll
      for (int ms = 0; ms < 2; ++ms)
#pragma unroll
        for (int ns = 0; ns < 4; ++ns)
          acc[ms][ns] = wmma4(af[ms], bf[ns], acc[ms][ns]);
    }
    wait_ds0();
    __syncthreads();
  }
#pragma unroll
  for (int ms = 0; ms < 2; ++ms)
#pragma unroll
    for (int ns = 0; ns < 4; ++ns)
#pragma unroll
      for (int i = 0; i < 8; ++i) {
        int row = t0 + wm * 32 + ms * 16 + i + (hi << 3);
        int col = n0 + wn * 64 + ns * 16 + m16;
        size_t idx = (size_t)row * Hh + col;
        out[idx] = out[idx] + acc[ms][ns][i];   // out already holds LoRA term
      }
}

// ---------------------------------------------------------------------------
extern "C" void kernel_launch(void* const* d_in, const int* in_sizes, int n_in,
                              void* d_out, int out_size, void* d_ws, size_t ws_size,
                              hipStream_t stream) {
  const float* x  = (const float*)d_in[0];   // [T,H]
  const float* rw = (const float*)d_in[1];   // [E,H]
  const float* rb = (const float*)d_in[2];   // [E]
  const float* w1 = (const float*)d_in[3];   // [FF,H]
  const float* w3 = (const float*)d_in[4];   // [FF,H]
  const float* w2 = (const float*)d_in[5];   // [H,FF]
  const float* lA = (const float*)d_in[6];   // [E,R,H]
  const float* lB = (const float*)d_in[7];   // [E,H,R]

  float* out    = (float*)d_out;                    // [T,H]
  float* logits = out + (size_t)Tt * Hh;            // [T,E]

  float* act    = (float*)d_ws;                     // [T,FF]  (~235 MB)
  float* zbuf   = act + (size_t)Tt * FFf;           // [T,32]
  int*   selbuf = (int*)(zbuf + (size_t)Tt * 32);   // [T]

  router_kernel  <<<Tt / 8, 256, 0, stream>>>(x, rw, rb, logits, selbuf);
  lora_z_kernel  <<<Tt / 8, 256, 0, stream>>>(x, lA, selbuf, zbuf);
  lora_out_kernel<<<(Tt * (Hh / 256)), 256, 0, stream>>>(zbuf, lB, selbuf, out);
  gate_up_kernel <<<dim3(FFf / 128, Tt / 128), 256, 0, stream>>>(x, w1, w3, act);
  down_kernel    <<<dim3(Hh / 128, Tt / 128), 256, 0, stream>>>(act, w2, out);
}